// RDB_10926396801884
// MI455X (gfx1250) — compile-verified
//
#include <hip/hip_runtime.h>
#include <math.h>

typedef float v2f __attribute__((ext_vector_type(2)));
typedef float v8f __attribute__((ext_vector_type(8)));

#define OMEGA 100.0f
#define BETA  0.2f

__device__ __forceinline__ float selu_f(float x) {
  const float sc = 1.0507009873554805f, al = 1.6732632423543772f;
  return x > 0.f ? sc * x : sc * al * (__expf(x) - 1.f);
}

// ---------------- edge spherical features (shared by all 3 layers) -------------
__global__ void edge_feat_kernel(const int* __restrict__ ei,
                                 const float* __restrict__ pos,
                                 float* __restrict__ feat, int E, int N) {
  int e = blockIdx.x * blockDim.x + threadIdx.x;
  int Etot = E + N;
  if (e >= Etot) return;
  int s, d;
  if (e < E) { s = ei[e]; d = ei[E + e]; } else { s = e - E; d = s; }
  float rx = pos[3*d+0] - pos[3*s+0];
  float ry = pos[3*d+1] - pos[3*s+1];
  float rz = pos[3*d+2] - pos[3*s+2];
  float sq = rx*rx + ry*ry + rz*rz;
  float rho = 0.f, th = 0.f, ph = 0.f;
  if (sq > 0.f) {
    rho = sqrtf(sq);
    th  = atan2f(ry, rx);
    float t = fminf(1.f, fmaxf(-1.f, rz / rho));
    ph  = asinf(t);
  }
  const float INVPI = 0.3183098861837907f;
  feat[3*e+0] = rho;
  feat[3*e+1] = th * INVPI;
  feat[3*e+2] = ph * INVPI;
}

__global__ void init_acc_kernel(float* __restrict__ acc, const float* __restrict__ bias,
                                int total, int CO) {
  int i = blockIdx.x * blockDim.x + threadIdx.x;
  if (i < total) acc[i] = bias[i % CO];
}

// ---------------- main SIREN-kernel continuous conv ----------------------------
// One wave32 per 16-row tile of (edge, out-channel) rows.
// A-matrix (f32 16x4): lane -> row M, {vgpr j, lane-half hl} -> K = 4*kc + 2*hl + j
// B-matrix (f32 4x16): lane -> col N, {vgpr j, lane-half hl} -> K (mirrors A)
// C/D (f32 16x16): vgpr v + 8*hl -> row M, lane&15 -> col N
template <int COUT, int CIN>
__global__ __launch_bounds__(256) void siren_conv_kernel(
    const float* __restrict__ feat, const float* __restrict__ xin,
    const int* __restrict__ ei,
    const float* __restrict__ w0, const float* __restrict__ b0,
    const float* __restrict__ w1, const float* __restrict__ b1,
    const float* __restrict__ w2, const float* __restrict__ b2,
    const float* __restrict__ wl, const float* __restrict__ bl,
    float* __restrict__ acc, int E, int Etot) {
  constexpr int NW  = 8;          // waves per block
  constexpr int EPT = 16 / COUT;  // edges per 16-row tile

  __shared__ float s_w0[16][4];
  __shared__ float s_b0[16], s_b1[16], s_b2[16];
  __shared__ float s_w1[16][16], s_w2[16][16];
  __shared__ float s_wl[CIN][16];
  __shared__ float s_bl[CIN];
  __shared__ float s_tile[NW][16][17];   // wave-private transpose staging (pad vs banks)
  __shared__ float s_t[NW][EPT][16];     // t = Wl^T x_j per tile edge
  __shared__ float s_s[NW][EPT];         // s = bl . x_j per tile edge

  const int tid = threadIdx.x;
  for (int i = tid; i < 256; i += blockDim.x) {
    s_w1[i >> 4][i & 15] = w1[i];
    s_w2[i >> 4][i & 15] = w2[i];
  }
  for (int i = tid; i < 16; i += blockDim.x) {
    s_b0[i] = b0[i]; s_b1[i] = b1[i]; s_b2[i] = b2[i];
    s_w0[i][0] = w0[4*i+0]; s_w0[i][1] = w0[4*i+1];
    s_w0[i][2] = w0[4*i+2]; s_w0[i][3] = w0[4*i+3];
  }
  for (int i = tid; i < CIN * 16; i += blockDim.x) s_wl[i >> 4][i & 15] = wl[i];
  for (int i = tid; i < CIN;      i += blockDim.x) s_bl[i] = bl[i];
  __syncthreads();

  const int  wave = tid >> 5;
  const int  lane = tid & 31;
  const int  m    = lane & 15;   // row (A) / col N (B,C)
  const int  hl   = lane >> 4;   // lane half
  const long R    = (long)Etot * COUT;
  const long row0 = ((long)blockIdx.x * NW + wave) * 16;
  const long r    = row0 + m;
  const bool vrow = (r < R);
  const int  e = vrow ? (int)(r / COUT) : 0;
  const int  o = vrow ? (int)(r % COUT) : 0;
  const float f0 = feat[3*e+0], f1 = feat[3*e+1], f2 = feat[3*e+2];

  // cooperative per-edge projections: t[h] = sum_c x_j[c]*wl[c][h]; s = sum_c x_j[c]*bl[c]
  {
    const int e0 = (int)(row0 / COUT);
    if (lane < EPT * 16) {
      const int te  = lane >> 4;
      const int h   = lane & 15;
      const int eg0 = e0 + te;
      const int eg  = (eg0 < Etot) ? eg0 : 0;
      const int sj  = (eg < E) ? ei[eg] : (eg - E);
      const float* xr = xin + (long)sj * CIN;
      float tv = 0.f, sv = 0.f;
      #pragma unroll
      for (int c = 0; c < CIN; ++c) {
        float xv = xr[c];
        tv = fmaf(xv, s_wl[c][h], tv);
        sv = fmaf(xv, s_bl[c], sv);
      }
      s_t[wave][te][h] = tv;
      if (h == 0) s_s[wave][te] = sv;
    }
  }

  // stage 0: h0 = sin(OMEGA*(feat . w0[:, :3] + b0 + o*w0[:,3]))  -> A layout
  v2f a0[4];
  #pragma unroll
  for (int kc = 0; kc < 4; ++kc)
    #pragma unroll
    for (int j = 0; j < 2; ++j) {
      int h = 4*kc + 2*hl + j;
      float pre = fmaf(f0, s_w0[h][0], fmaf(f1, s_w0[h][1], fmaf(f2, s_w0[h][2], s_b0[h])));
      pre = fmaf((float)o, s_w0[h][3], pre);
      a0[kc][j] = __sinf(OMEGA * pre);
    }

  // W1^T, W2^T in B layout: B[k=h][n=g] = w[g][h]
  v2f bw1[4], bw2[4];
  #pragma unroll
  for (int kc = 0; kc < 4; ++kc)
    #pragma unroll
    for (int j = 0; j < 2; ++j) {
      int k = 4*kc + 2*hl + j;
      bw1[kc][j] = s_w1[m][k];
      bw2[kc][j] = s_w2[m][k];
    }

  // GEMM 1: D1[row, g] = sum_h h0[row, h] * w1[g, h]   (4x K-chunked fp32 WMMA)
  v8f c1 = {0.f,0.f,0.f,0.f,0.f,0.f,0.f,0.f};
  #pragma unroll
  for (int kc = 0; kc < 4; ++kc)
    c1 = __builtin_amdgcn_wmma_f32_16x16x4_f32(false, a0[kc], false, bw1[kc],
                                               (short)0, c1, false, false);

  // sin + transpose through wave-private LDS (same-wave DS ops are in-order)
  {
    float bb = s_b1[m];
    #pragma unroll
    for (int v = 0; v < 8; ++v)
      s_tile[wave][v + 8*hl][m] = __sinf(OMEGA * (c1[v] + bb));
  }
  v2f a1[4];
  #pragma unroll
  for (int kc = 0; kc < 4; ++kc)
    #pragma unroll
    for (int j = 0; j < 2; ++j)
      a1[kc][j] = s_tile[wave][m][4*kc + 2*hl + j];

  // GEMM 2
  v8f c2 = {0.f,0.f,0.f,0.f,0.f,0.f,0.f,0.f};
  #pragma unroll
  for (int kc = 0; kc < 4; ++kc)
    c2 = __builtin_amdgcn_wmma_f32_16x16x4_f32(false, a1[kc], false, bw2[kc],
                                               (short)0, c2, false, false);

  {
    float bb = s_b2[m];
    #pragma unroll
    for (int v = 0; v < 8; ++v)
      s_tile[wave][v + 8*hl][m] = __sinf(OMEGA * (c2[v] + bb));
  }

  // message: msg[row] = h2[row,:] . t[edge,:] + s[edge]; scatter-add onto dst node
  if (lane < 16 && vrow) {
    const int te2 = m / COUT;
    float msg = s_s[wave][te2];
    #pragma unroll
    for (int h = 0; h < 16; ++h)
      msg = fmaf(s_tile[wave][m][h], s_t[wave][te2][h], msg);
    const int dj = (e < E) ? ei[E + e] : (e - E);
    atomicAdd(&acc[(long)dj * COUT + o], msg);
  }
}

// ---------------- elementwise glue --------------------------------------------
__global__ void post1_kernel(const float* __restrict__ x, const float* __restrict__ acc1,
                             float* __restrict__ xc2, float* __restrict__ xc3, int N) {
  int i = blockIdx.x * blockDim.x + threadIdx.x;
  if (i >= N * 24) return;
  int n = i / 24, c = i % 24;
  float v = (c < 16) ? x[n*16 + c] : selu_f(acc1[n*8 + (c - 16)]);
  xc2[i] = v;
  xc3[n*32 + c] = v;
}

__global__ void post2_kernel(const float* __restrict__ acc2, float* __restrict__ xc3, int N) {
  int i = blockIdx.x * blockDim.x + threadIdx.x;
  if (i >= N * 8) return;
  int n = i / 8, c = i % 8;
  xc3[n*32 + 24 + c] = selu_f(acc2[i]);
}

__global__ void final_kernel(const float* __restrict__ x, const float* __restrict__ acc3,
                             float* __restrict__ out, int N) {
  int i = blockIdx.x * blockDim.x + threadIdx.x;
  if (i >= N * 16) return;
  out[i] = x[i] + BETA * selu_f(acc3[i]);
}

// ---------------- launch -------------------------------------------------------
extern "C" void kernel_launch(void* const* d_in, const int* in_sizes, int n_in,
                              void* d_out, int out_size, void* d_ws, size_t ws_size,
                              hipStream_t stream) {
  (void)n_in; (void)out_size; (void)ws_size;
  const float* x   = (const float*)d_in[0];
  const int*   ei  = (const int*)d_in[1];
  const float* pos = (const float*)d_in[2];
  const int N    = in_sizes[0] / 16;
  const int E    = in_sizes[1] / 2;
  const int Etot = E + N;

  auto F = [&](int i) { return (const float*)d_in[i]; };

  float* ws   = (float*)d_ws;
  float* feat = ws;                          // 3*Etot
  float* acc1 = feat + (size_t)3 * Etot;     // N*8
  float* acc2 = acc1 + (size_t)N * 8;        // N*8
  float* acc3 = acc2 + (size_t)N * 8;        // N*16
  float* xc2  = acc3 + (size_t)N * 16;       // N*24
  float* xc3  = xc2  + (size_t)N * 24;       // N*32

  const int T = 256;
  edge_feat_kernel<<<(Etot + T - 1) / T, T, 0, stream>>>(ei, pos, feat, E, N);

  long R8  = (long)Etot * 8;
  long R16 = (long)Etot * 16;
  int tiles8  = (int)((R8  + 15) / 16);
  int tiles16 = (int)((R16 + 15) / 16);
  int blocks8  = (tiles8  + 7) / 8;
  int blocks16 = (tiles16 + 7) / 8;

  // Layer 1: CIN=16 -> COUT=8
  init_acc_kernel<<<(N*8 + T - 1)/T, T, 0, stream>>>(acc1, F(11), N*8, 8);
  siren_conv_kernel<8,16><<<blocks8, T, 0, stream>>>(feat, x, ei,
      F(3), F(4), F(5), F(6), F(7), F(8), F(9), F(10), acc1, E, Etot);
  post1_kernel<<<(N*24 + T - 1)/T, T, 0, stream>>>(x, acc1, xc2, xc3, N);

  // Layer 2: CIN=24 -> COUT=8
  init_acc_kernel<<<(N*8 + T - 1)/T, T, 0, stream>>>(acc2, F(20), N*8, 8);
  siren_conv_kernel<8,24><<<blocks8, T, 0, stream>>>(feat, xc2, ei,
      F(12), F(13), F(14), F(15), F(16), F(17), F(18), F(19), acc2, E, Etot);
  post2_kernel<<<(N*8 + T - 1)/T, T, 0, stream>>>(acc2, xc3, N);

  // Layer 3: CIN=32 -> COUT=16
  init_acc_kernel<<<(N*16 + T - 1)/T, T, 0, stream>>>(acc3, F(29), N*16, 16);
  siren_conv_kernel<16,32><<<blocks16, T, 0, stream>>>(feat, xc3, ei,
      F(21), F(22), F(23), F(24), F(25), F(26), F(27), F(28), acc3, E, Etot);
  final_kernel<<<(N*16 + T - 1)/T, T, 0, stream>>>(x, acc3, (float*)d_out, N);
}